// DimeNetBlock_37220186587473
// MI455X (gfx1250) — compile-verified
//
#include <hip/hip_runtime.h>
#include <math.h>

typedef float v2f __attribute__((ext_vector_type(2)));
typedef float v8f __attribute__((ext_vector_type(8)));

// D = A(16x4 f32) * B(4x16 f32) + C(16x16 f32), wave32 CDNA5 WMMA
#define WMMA_F32(a, b, c) \
  __builtin_amdgcn_wmma_f32_16x16x4_f32(false, (a), false, (b), (short)0, (c), false, false)

__device__ __forceinline__ float silu_f(float h) {
  return h / (1.0f + __expf(-h));
}

// ---------------------------------------------------------------------------
// Fused edge kernel: angle + edge MLP (17->64 SiLU 64->64) + atomic scatter-add
// One wave = one 16-edge tile. 8 waves / block.
// ---------------------------------------------------------------------------
__global__ __launch_bounds__(256) void dimenet_edge_kernel(
    const float* __restrict__ coord,        // [N,3]
    const float* __restrict__ rbf,          // [E,16]
    const long long* __restrict__ edge_index, // [2,E] int64
    const float* __restrict__ W1, const float* __restrict__ b1, // [17,64],[64]
    const float* __restrict__ W2, const float* __restrict__ b2, // [64,64],[64]
    float* __restrict__ agg,                // [N,64] (pre-zeroed)
    long long E_, long long N_)
{
  __shared__ float w1T[64 * 20];            // W1 transposed [n][k], k padded 17->20
  __shared__ float w2T[64 * 68];            // W2 transposed [n][k], stride 68 (bank-safe)
  __shared__ float b1s[64];
  __shared__ float b2s[64];
  __shared__ float stageBuf[8][16 * 68];    // per-wave activation transpose tile
  __shared__ int   rowBuf[8][16];           // per-wave destination rows

  const int tid = threadIdx.x;
  for (int i = tid; i < 64 * 20; i += 256) {
    int n = i / 20, k = i % 20;
    w1T[i] = (k < 17) ? W1[k * 64 + n] : 0.0f;
  }
  for (int i = tid; i < 64 * 68; i += 256) {
    int n = i / 68, k = i % 68;
    w2T[i] = (k < 64) ? W2[k * 64 + n] : 0.0f;
  }
  if (tid < 64) { b1s[tid] = b1[tid]; b2s[tid] = b2[tid]; }
  __syncthreads();

  const int wave = tid >> 5;
  const int lane = tid & 31;
  const int half = lane >> 4;   // 0: K low pair, 1: K high pair (A/B frag layout)
  const int lm   = lane & 15;
  const int kh   = 2 * half;
  float* st = stageBuf[wave];
  int*   rI = rowBuf[wave];

  const long long numTiles = (E_ + 15) >> 4;
  const long long tile = (long long)blockIdx.x * 8 + wave;
  if (tile >= numTiles) return;             // wave-uniform: EXEC stays all-1

  const long long ebase = tile << 4;
  long long e = ebase + lm;
  if (e >= E_) e = E_ - 1;                  // clamp duplicate for tail tile

  const long long r = edge_index[e];
  const long long c = edge_index[E_ + e];
  if (half == 0) rI[lm] = (int)r;

  // angle(v, -v) computed faithfully in fp32
  const float vx = coord[r * 3 + 0] - coord[c * 3 + 0];
  const float vy = coord[r * 3 + 1] - coord[c * 3 + 1];
  const float vz = coord[r * 3 + 2] - coord[c * 3 + 2];
  float nrm = sqrtf(vx * vx + vy * vy + vz * vz);
  nrm = fmaxf(nrm, 1e-12f);
  const float ux = vx / nrm, uy = vy / nrm, uz = vz / nrm;
  float cosang = -(ux * ux + uy * uy + uz * uz);
  cosang = fminf(fmaxf(cosang, -1.0f + 1e-8f), 1.0f - 1e-8f);
  const float angle = acosf(cosang);

  // A fragments, layer 1: feat[16 x 20] (16 rbf + angle + zero pad)
  v2f a1[5];
#pragma unroll
  for (int s = 0; s < 4; ++s)
    a1[s] = *reinterpret_cast<const v2f*>(rbf + e * 16 + 4 * s + kh);
  a1[4] = (half == 0) ? (v2f){angle, 0.0f} : (v2f){0.0f, 0.0f};

  // layer 1: h1[16x64] = feat @ W1   (5 k-steps x 4 n-tiles)
  v8f acc[4];
#pragma unroll
  for (int t = 0; t < 4; ++t)
    acc[t] = (v8f){0.f, 0.f, 0.f, 0.f, 0.f, 0.f, 0.f, 0.f};
#pragma unroll
  for (int t = 0; t < 4; ++t) {
    const float* wp = w1T + (t * 16 + lm) * 20 + kh;
#pragma unroll
    for (int s = 0; s < 5; ++s) {
      v2f b = *reinterpret_cast<const v2f*>(wp + 4 * s);
      acc[t] = WMMA_F32(a1[s], b, acc[t]);
    }
  }

  // bias + SiLU, write C-layout -> padded LDS tile (conflict-free, stride 68)
#pragma unroll
  for (int t = 0; t < 4; ++t) {
    const float bias = b1s[t * 16 + lm];
#pragma unroll
    for (int j = 0; j < 8; ++j) {
      const int M = j + 8 * half;
      st[M * 68 + t * 16 + lm] = silu_f(acc[t][j] + bias);
    }
  }
  asm volatile("s_wait_dscnt 0" ::: "memory");  // intra-wave LDS store->load order

  // re-read as A-layout for layer 2
  v2f a2[16];
#pragma unroll
  for (int s = 0; s < 16; ++s)
    a2[s] = *reinterpret_cast<const v2f*>(st + lm * 68 + 4 * s + kh);

  // layer 2: msg[16x64] = act @ W2  (16 k-steps x 4 n-tiles)
  v8f acc2[4];
#pragma unroll
  for (int t = 0; t < 4; ++t)
    acc2[t] = (v8f){0.f, 0.f, 0.f, 0.f, 0.f, 0.f, 0.f, 0.f};
#pragma unroll
  for (int t = 0; t < 4; ++t) {
    const float* wp = w2T + (t * 16 + lm) * 68 + kh;
#pragma unroll
    for (int s = 0; s < 16; ++s) {
      v2f b = *reinterpret_cast<const v2f*>(wp + 4 * s);
      acc2[t] = WMMA_F32(a2[s], b, acc2[t]);
    }
  }

  // bias + scatter-add into agg (L2-resident, 25.6 MB)
#pragma unroll
  for (int t = 0; t < 4; ++t) {
    const float bias = b2s[t * 16 + lm];
#pragma unroll
    for (int j = 0; j < 8; ++j) {
      const int M = j + 8 * half;
      if (ebase + M < E_) {
        const long long rowIdx = rI[M];
        atomicAdd(agg + rowIdx * 64 + t * 16 + lm, acc2[t][j] + bias);
      }
    }
  }
}

// ---------------------------------------------------------------------------
// Node kernel: out = x + W4(SiLU(W3(agg)+b3))+b4 ; one wave = 16 nodes.
// 4 waves / block (LDS budget).
// ---------------------------------------------------------------------------
__global__ __launch_bounds__(128) void dimenet_node_kernel(
    const float* __restrict__ x,            // [N,64]
    const float* __restrict__ agg,          // [N,64]
    const float* __restrict__ W3, const float* __restrict__ b3,
    const float* __restrict__ W4, const float* __restrict__ b4,
    float* __restrict__ out,                // [N,64]
    long long N_)
{
  __shared__ float w3T[64 * 68];
  __shared__ float w4T[64 * 68];
  __shared__ float b3s[64];
  __shared__ float b4s[64];
  __shared__ float stageBuf[4][16 * 68];

  const int tid = threadIdx.x;
  for (int i = tid; i < 64 * 68; i += 128) {
    int n = i / 68, k = i % 68;
    float v3 = (k < 64) ? W3[k * 64 + n] : 0.0f;
    float v4 = (k < 64) ? W4[k * 64 + n] : 0.0f;
    w3T[i] = v3;
    w4T[i] = v4;
  }
  if (tid < 64) { b3s[tid] = b3[tid]; b4s[tid] = b4[tid]; }
  __syncthreads();

  const int wave = tid >> 5;
  const int lane = tid & 31;
  const int half = lane >> 4;
  const int lm   = lane & 15;
  const int kh   = 2 * half;
  float* st = stageBuf[wave];

  const long long numTiles = (N_ + 15) >> 4;
  const long long tile = (long long)blockIdx.x * 4 + wave;
  if (tile >= numTiles) return;
  const long long nbase = tile << 4;

  long long node = nbase + lm;
  if (node >= N_) node = N_ - 1;

  // A fragments: agg rows are contiguous -> direct global b64 loads
  v2f a1[16];
#pragma unroll
  for (int s = 0; s < 16; ++s)
    a1[s] = *reinterpret_cast<const v2f*>(agg + node * 64 + 4 * s + kh);

  v8f acc[4];
#pragma unroll
  for (int t = 0; t < 4; ++t)
    acc[t] = (v8f){0.f, 0.f, 0.f, 0.f, 0.f, 0.f, 0.f, 0.f};
#pragma unroll
  for (int t = 0; t < 4; ++t) {
    const float* wp = w3T + (t * 16 + lm) * 68 + kh;
#pragma unroll
    for (int s = 0; s < 16; ++s) {
      v2f b = *reinterpret_cast<const v2f*>(wp + 4 * s);
      acc[t] = WMMA_F32(a1[s], b, acc[t]);
    }
  }

#pragma unroll
  for (int t = 0; t < 4; ++t) {
    const float bias = b3s[t * 16 + lm];
#pragma unroll
    for (int j = 0; j < 8; ++j) {
      const int M = j + 8 * half;
      st[M * 68 + t * 16 + lm] = silu_f(acc[t][j] + bias);
    }
  }
  asm volatile("s_wait_dscnt 0" ::: "memory");

  v2f a2[16];
#pragma unroll
  for (int s = 0; s < 16; ++s)
    a2[s] = *reinterpret_cast<const v2f*>(st + lm * 68 + 4 * s + kh);

  v8f acc2[4];
#pragma unroll
  for (int t = 0; t < 4; ++t)
    acc2[t] = (v8f){0.f, 0.f, 0.f, 0.f, 0.f, 0.f, 0.f, 0.f};
#pragma unroll
  for (int t = 0; t < 4; ++t) {
    const float* wp = w4T + (t * 16 + lm) * 68 + kh;
#pragma unroll
    for (int s = 0; s < 16; ++s) {
      v2f b = *reinterpret_cast<const v2f*>(wp + 4 * s);
      acc2[t] = WMMA_F32(a2[s], b, acc2[t]);
    }
  }

  // residual + bias + store
#pragma unroll
  for (int t = 0; t < 4; ++t) {
    const float bias = b4s[t * 16 + lm];
#pragma unroll
    for (int j = 0; j < 8; ++j) {
      const int M = j + 8 * half;
      const long long nd = nbase + M;
      if (nd < N_) {
        const long long idx = nd * 64 + t * 16 + lm;
        out[idx] = x[idx] + acc2[t][j] + bias;
      }
    }
  }
}

// ---------------------------------------------------------------------------
extern "C" void kernel_launch(void* const* d_in, const int* in_sizes, int n_in,
                              void* d_out, int out_size, void* d_ws, size_t ws_size,
                              hipStream_t stream) {
  const float* x        = (const float*)d_in[0];
  const float* coord    = (const float*)d_in[1];
  const float* rbf      = (const float*)d_in[2];
  const long long* ei   = (const long long*)d_in[3];   // int64 per reference
  const float* W1 = (const float*)d_in[4];
  const float* b1 = (const float*)d_in[5];
  const float* W2 = (const float*)d_in[6];
  const float* b2 = (const float*)d_in[7];
  const float* W3 = (const float*)d_in[8];
  const float* b3 = (const float*)d_in[9];
  const float* W4 = (const float*)d_in[10];
  const float* b4 = (const float*)d_in[11];
  float* out = (float*)d_out;

  const long long N_ = (long long)in_sizes[0] / 64;   // x:   [N,64]
  const long long E_ = (long long)in_sizes[2] / 16;   // rbf: [E,16]

  float* agg = (float*)d_ws;                          // [N,64] scratch
  hipMemsetAsync(agg, 0, (size_t)N_ * 64 * sizeof(float), stream);

  const long long tilesE = (E_ + 15) >> 4;
  const int blocksE = (int)((tilesE + 7) / 8);
  dimenet_edge_kernel<<<blocksE, 256, 0, stream>>>(coord, rbf, ei, W1, b1, W2, b2,
                                                   agg, E_, N_);

  const long long tilesN = (N_ + 15) >> 4;
  const int blocksN = (int)((tilesN + 3) / 4);
  dimenet_node_kernel<<<blocksN, 128, 0, stream>>>(x, agg, W3, b3, W4, b4, out, N_);
}